// GraphCBAM_58299886076478
// MI455X (gfx1250) — compile-verified
//
#include <hip/hip_runtime.h>
#include <math.h>

#define CDIM 256
#define HID  32          // C / R
#define NB_RED 1024      // partial-reduction blocks

typedef __attribute__((ext_vector_type(2))) float v2f;
typedef __attribute__((ext_vector_type(4))) float v4f;
typedef __attribute__((ext_vector_type(8))) float v8f;

#if __has_builtin(__builtin_amdgcn_wmma_f32_16x16x4_f32)
#define HAVE_WMMA4 1
#else
#define HAVE_WMMA4 0
#endif

__device__ __forceinline__ float wave_sum32(float v) {
#pragma unroll
  for (int o = 16; o > 0; o >>= 1) v += __shfl_xor(v, o, 32);
  return v;
}
__device__ __forceinline__ float wave_max32(float v) {
#pragma unroll
  for (int o = 16; o > 0; o >>= 1) v = fmaxf(v, __shfl_xor(v, o, 32));
  return v;
}

// ---------------------------------------------------------------------------
// Kernel 1: partial column reduction (sum + max over nodes, per channel).
// 256 threads = 4 row-groups x 64 col-quads; each thread covers 4 channels.
// ---------------------------------------------------------------------------
__global__ __launch_bounds__(256) void k_col_partial(
    const float* __restrict__ x, float* __restrict__ psum,
    float* __restrict__ pmax, int N)
{
  const int tid = threadIdx.x;
  const int g = tid >> 6;      // row group 0..3
  const int t = tid & 63;      // column quad: channels 4t..4t+3
  const int b = blockIdx.x;
  const int rows_per_blk = (N + NB_RED - 1) / NB_RED;
  const int r0 = b * rows_per_blk;
  const int r1 = (r0 + rows_per_blk < N) ? (r0 + rows_per_blk) : N;

  float sx = 0.f, sy = 0.f, sz = 0.f, sw = 0.f;
  float mxx = -INFINITY, mxy = -INFINITY, mxz = -INFINITY, mxw = -INFINITY;
  for (int n = r0 + g; n < r1; n += 4) {
    const v4f v = __builtin_nontemporal_load(
        reinterpret_cast<const v4f*>(x + (size_t)n * CDIM) + t);
    sx += v.x; sy += v.y; sz += v.z; sw += v.w;
    mxx = fmaxf(mxx, v.x); mxy = fmaxf(mxy, v.y);
    mxz = fmaxf(mxz, v.z); mxw = fmaxf(mxw, v.w);
  }

  __shared__ float s_ps[4][CDIM];
  __shared__ float s_pm[4][CDIM];
  s_ps[g][4*t+0] = sx; s_ps[g][4*t+1] = sy; s_ps[g][4*t+2] = sz; s_ps[g][4*t+3] = sw;
  s_pm[g][4*t+0] = mxx; s_pm[g][4*t+1] = mxy; s_pm[g][4*t+2] = mxz; s_pm[g][4*t+3] = mxw;
  __syncthreads();

  const int c = tid; // 256 threads == CDIM channels
  const float ss = s_ps[0][c] + s_ps[1][c] + s_ps[2][c] + s_ps[3][c];
  const float mm = fmaxf(fmaxf(s_pm[0][c], s_pm[1][c]), fmaxf(s_pm[2][c], s_pm[3][c]));
  psum[(size_t)b * CDIM + c] = ss;
  pmax[(size_t)b * CDIM + c] = mm;
}

// ---------------------------------------------------------------------------
// Kernel 2: finalize column reduction, then channel-attention MLP via WMMA.
//   att_c = sigmoid( W2*relu(W1*avg) + W2*relu(W1*max) )
// The 256x16 B operand (cols 0=avg, 1=max, 2..15=zero) is staged zero-padded
// in LDS so every lane's fragment load is a plain address computation ->
// unconditional ds_load, no selects, no EXEC save/restore in the WMMA loops.
// ---------------------------------------------------------------------------
__global__ __launch_bounds__(1024) void k_attc(
    const float* __restrict__ psum, const float* __restrict__ pmax,
    const float* __restrict__ w1, const float* __restrict__ w2,
    float* __restrict__ att_c, int N)
{
  __shared__ float s_w1[HID * CDIM];    // [h][c] row-major, 32 KB
  __shared__ float s_w2[CDIM * HID];    // [c][h] row-major, 32 KB
  __shared__ float s_vcols[CDIM * 16];  // B staging stage A: [k][ncol], 16 KB
  __shared__ float s_hcols[HID * 16];   // B staging stage B: [k][ncol], 2 KB
  __shared__ float s_o0[CDIM], s_o1[CDIM];
  __shared__ float s_rs[4][CDIM], s_rm[4][CDIM];

  const int tid = threadIdx.x;
  const int grp = tid >> 8;   // 0..3
  const int c = tid & 255;

  float s = 0.f, mx = -INFINITY;
  for (int b = grp; b < NB_RED; b += 4) {
    s += psum[(size_t)b * CDIM + c];
    mx = fmaxf(mx, pmax[(size_t)b * CDIM + c]);
  }
  s_rs[grp][c] = s; s_rm[grp][c] = mx;

  for (int i = tid; i < HID * CDIM; i += 1024) s_w1[i] = w1[i];
  for (int i = tid; i < CDIM * HID; i += 1024) s_w2[i] = w2[i];
  for (int i = tid; i < CDIM * 16; i += 1024) s_vcols[i] = 0.f;
  for (int i = tid; i < HID * 16; i += 1024) s_hcols[i] = 0.f;
  __syncthreads();

  if (tid < CDIM) {
    const float avg = (s_rs[0][tid] + s_rs[1][tid] + s_rs[2][tid] + s_rs[3][tid])
                      / (float)N;
    const float mxv = fmaxf(fmaxf(s_rm[0][tid], s_rm[1][tid]),
                            fmaxf(s_rm[2][tid], s_rm[3][tid]));
    s_vcols[tid * 16 + 0] = avg;
    s_vcols[tid * 16 + 1] = mxv;
  }
  __syncthreads();

#if HAVE_WMMA4
  if (tid < 32) {             // wave 0 only: EXEC all-1s within the wave
    const int lane  = tid;
    const int mrow  = lane & 15;              // A row / D column index
    const int khalf = (lane < 16) ? 0 : 2;    // A/B K sub-block per lane half
    const int ncol  = mrow;                   // B/D column
    // ---- Stage A: H(32x16) = W1(32x256) x V(256x16), cols 0/1 = avg/max ----
    for (int tile = 0; tile < 2; ++tile) {
      v8f acc = {};
      for (int k0 = 0; k0 < CDIM; k0 += 4) {
        const int k = k0 + khalf;
        v2f A, B;
        A.x = s_w1[(tile * 16 + mrow) * CDIM + k];
        A.y = s_w1[(tile * 16 + mrow) * CDIM + k + 1];
        B.x = s_vcols[k * 16 + ncol];
        B.y = s_vcols[(k + 1) * 16 + ncol];
        acc = __builtin_amdgcn_wmma_f32_16x16x4_f32(
            false, A, false, B, (short)0, acc, false, false);
      }
      if (ncol < 2) {
#pragma unroll
        for (int v = 0; v < 8; ++v) {
          const int hrow = tile * 16 + v + ((lane < 16) ? 0 : 8);
          s_hcols[hrow * 16 + ncol] = fmaxf(acc[v], 0.f);   // ReLU
        }
      }
    }
    // ---- Stage B: O(256x16) = W2(256x32) x H(32x16) ----
    for (int tile = 0; tile < 16; ++tile) {
      v8f acc = {};
#pragma unroll
      for (int k0 = 0; k0 < HID; k0 += 4) {
        const int k = k0 + khalf;
        v2f A, B;
        A.x = s_w2[(tile * 16 + mrow) * HID + k];
        A.y = s_w2[(tile * 16 + mrow) * HID + k + 1];
        B.x = s_hcols[k * 16 + ncol];
        B.y = s_hcols[(k + 1) * 16 + ncol];
        acc = __builtin_amdgcn_wmma_f32_16x16x4_f32(
            false, A, false, B, (short)0, acc, false, false);
      }
#pragma unroll
      for (int v = 0; v < 8; ++v) {
        const int row = tile * 16 + v + ((lane < 16) ? 0 : 8);
        if (ncol == 0) s_o0[row] = acc[v];
        if (ncol == 1) s_o1[row] = acc[v];
      }
    }
  }
  __syncthreads();
#else
  // Scalar fallback (host pass / missing builtin)
  if (tid < 2 * HID) {
    const int path = tid / HID, k = tid % HID;
    float acc = 0.f;
    for (int cc = 0; cc < CDIM; ++cc)
      acc += s_w1[k * CDIM + cc] * s_vcols[cc * 16 + path];
    s_hcols[k * 16 + path] = fmaxf(acc, 0.f);
  }
  __syncthreads();
  if (tid < CDIM) {
    float o0 = 0.f, o1 = 0.f;
    for (int k = 0; k < HID; ++k) {
      o0 += s_w2[tid * HID + k] * s_hcols[k * 16 + 0];
      o1 += s_w2[tid * HID + k] * s_hcols[k * 16 + 1];
    }
    s_o0[tid] = o0; s_o1[tid] = o1;
  }
  __syncthreads();
#endif

  if (tid < CDIM) {
    const float z = s_o0[tid] + s_o1[tid];
    att_c[tid] = 1.f / (1.f + expf(-z));
  }
}

// ---------------------------------------------------------------------------
// Kernel 3: fused  xc = x*att_c -> LayerNorm -> per-row mean/max ->
//           width-3 conv over nodes -> sigmoid gate -> out = xc*(1+gamma*s).
// One wave per row; 32 rows per block; waves 0/31 recompute +/-1 halo rows.
// ---------------------------------------------------------------------------
__device__ __forceinline__ void row_stats(
    const float* __restrict__ xrow, const float* s_att, const float* s_g,
    const float* s_b, int lane, float* xr, float& a_out, float& m_out)
{
  const v4f* xp = reinterpret_cast<const v4f*>(xrow);
  const v4f v0 = __builtin_nontemporal_load(xp + lane);
  const v4f v1 = __builtin_nontemporal_load(xp + lane + 32);
  const int c0 = lane * 4;
  const int c1 = lane * 4 + 128;
  xr[0] = v0.x * s_att[c0 + 0]; xr[1] = v0.y * s_att[c0 + 1];
  xr[2] = v0.z * s_att[c0 + 2]; xr[3] = v0.w * s_att[c0 + 3];
  xr[4] = v1.x * s_att[c1 + 0]; xr[5] = v1.y * s_att[c1 + 1];
  xr[6] = v1.z * s_att[c1 + 2]; xr[7] = v1.w * s_att[c1 + 3];

  float s = 0.f, sq = 0.f;
#pragma unroll
  for (int j = 0; j < 8; ++j) { s += xr[j]; sq += xr[j] * xr[j]; }
  s = wave_sum32(s); sq = wave_sum32(sq);
  const float mu = s * (1.f / CDIM);
  float var = sq * (1.f / CDIM) - mu * mu;
  var = fmaxf(var, 0.f);
  const float rstd = rsqrtf(var + 1e-5f);

  float asum = 0.f, amax = -INFINITY;
#pragma unroll
  for (int j = 0; j < 8; ++j) {
    const int c = (j < 4) ? (c0 + j) : (c1 + j - 4);
    const float xn = (xr[j] - mu) * rstd * s_g[c] + s_b[c];
    asum += xn; amax = fmaxf(amax, xn);
  }
  a_out = wave_sum32(asum) * (1.f / CDIM);
  m_out = wave_max32(amax);
}

__global__ __launch_bounds__(1024) void k_ln_conv_out(
    const float* __restrict__ x, const float* __restrict__ att,
    const float* __restrict__ ln_g, const float* __restrict__ ln_b,
    const float* __restrict__ conv_w, const float* __restrict__ gamma,
    float* __restrict__ out, int N)
{
  __shared__ float s_att[CDIM], s_g[CDIM], s_b[CDIM];
  __shared__ float s_a[34], s_m[34];
  __shared__ float s_cw[6];
  __shared__ float s_gamma;

  const int tid = threadIdx.x;
  const int wave = tid >> 5;
  const int lane = tid & 31;
  if (tid < CDIM) { s_att[tid] = att[tid]; s_g[tid] = ln_g[tid]; s_b[tid] = ln_b[tid]; }
  if (tid == CDIM) s_gamma = gamma[0];
  if (tid > CDIM && tid <= CDIM + 6) s_cw[tid - CDIM - 1] = conv_w[tid - CDIM - 1];
  __syncthreads();

  const int base = blockIdx.x * 32;
  const int n = base + wave;

  float xr[8];
  float a, m;
  if (n < N) {
    row_stats(x + (size_t)n * CDIM, s_att, s_g, s_b, lane, xr, a, m);
    if (lane == 0) { s_a[wave + 1] = a; s_m[wave + 1] = m; }
  } else if (lane == 0) {
    s_a[wave + 1] = 0.f; s_m[wave + 1] = 0.f;   // zero-pad past end
  }

  if (wave == 0) {          // left halo row base-1
    float ha = 0.f, hm = 0.f, hx[8];
    const int hn = base - 1;
    if (hn >= 0) row_stats(x + (size_t)hn * CDIM, s_att, s_g, s_b, lane, hx, ha, hm);
    if (lane == 0) { s_a[0] = ha; s_m[0] = hm; }
  }
  if (wave == 31) {         // right halo row base+32
    float ha = 0.f, hm = 0.f, hx[8];
    const int hn = base + 32;
    if (hn < N) row_stats(x + (size_t)hn * CDIM, s_att, s_g, s_b, lane, hx, ha, hm);
    if (lane == 0) { s_a[33] = ha; s_m[33] = hm; }
  }
  __syncthreads();

  if (n < N) {
    const float z = s_cw[0] * s_a[wave]     + s_cw[1] * s_a[wave + 1]
                  + s_cw[2] * s_a[wave + 2] + s_cw[3] * s_m[wave]
                  + s_cw[4] * s_m[wave + 1] + s_cw[5] * s_m[wave + 2];
    const float sig = 1.f / (1.f + expf(-z));
    const float scale = 1.f + s_gamma * sig;
    v4f o0, o1;
    o0.x = xr[0] * scale; o0.y = xr[1] * scale;
    o0.z = xr[2] * scale; o0.w = xr[3] * scale;
    o1.x = xr[4] * scale; o1.y = xr[5] * scale;
    o1.z = xr[6] * scale; o1.w = xr[7] * scale;
    v4f* op = reinterpret_cast<v4f*>(out + (size_t)n * CDIM);
    __builtin_nontemporal_store(o0, op + lane);
    __builtin_nontemporal_store(o1, op + lane + 32);
  }
}

// ---------------------------------------------------------------------------
extern "C" void kernel_launch(void* const* d_in, const int* in_sizes, int n_in,
                              void* d_out, int out_size, void* d_ws, size_t ws_size,
                              hipStream_t stream)
{
  const float* x      = (const float*)d_in[0];
  const float* w1     = (const float*)d_in[1];
  const float* w2     = (const float*)d_in[2];
  const float* conv_w = (const float*)d_in[3];
  const float* ln_g   = (const float*)d_in[4];
  const float* ln_b   = (const float*)d_in[5];
  const float* gamma  = (const float*)d_in[6];
  float* out = (float*)d_out;
  const int N = in_sizes[0] / CDIM;

  float* psum = (float*)d_ws;                         // NB_RED * 256
  float* pmax = psum + (size_t)NB_RED * CDIM;         // NB_RED * 256
  float* attc = pmax + (size_t)NB_RED * CDIM;         // 256

  k_col_partial<<<NB_RED, 256, 0, stream>>>(x, psum, pmax, N);
  k_attc<<<1, 1024, 0, stream>>>(psum, pmax, w1, w2, attc, N);
  const int nblk = (N + 31) / 32;
  k_ln_conv_out<<<nblk, 1024, 0, stream>>>(x, attc, ln_g, ln_b, conv_w, gamma, out, N);
}